// ConvDropoutNormReLU_2465311228182
// MI455X (gfx1250) — compile-verified
//
#include <hip/hip_runtime.h>
#include <hip/hip_bf16.h>
#include <math.h>

typedef __attribute__((ext_vector_type(2))) float v2f;
typedef __attribute__((ext_vector_type(8))) float v8f;

#define BATCH 2
#define S 80
#define VOX (S*S*S)            // 512000 per batch
#define CIN 4
#define CMID 72                // 28 scalar + 24 (l=1) + 20 (l=2)
#define CPAD 80
#define KTOT 108               // CIN * 27
#define COUT 60
#define NS_ 16
#define NG_ 12
#define M1_ 8
#define M2_ 4
#define EPSV 1e-5f
#define RELU_C 1.41421356237309515f
#define SIG_C 1.8464f

// ---- workspace layout (floats) — tiny now, no materialized h ----
#define OFF_BMAT  0
#define OFF_SUM   (OFF_BMAT + (size_t)KTOT*CPAD)   // 8640
#define OFF_SS    (OFF_SUM + BATCH*CMID)
#define OFF_SC    (OFF_SS  + BATCH*CMID)
#define OFF_SH    (OFF_SC  + BATCH*CMID)

// A-matrix LDS offset for GEMM index k (folds to a constant after unroll):
// k = ci*27 + p ; p = dz*9 + dy*3 + dx ; slab layout xs[ci][dz][dy][82]
__device__ __forceinline__ constexpr int aoffs(int k) {
    int ci = k / 27, p = k % 27;
    int dz = p / 9, dy = (p % 9) / 3, dx = p % 3;
    return ((ci * 3 + dz) * 3 + dy) * 82 + dx;
}

// ---------------- kernel 0: build B-matrix (108 x 80) + zero stats -------------
__global__ void build_kernel_k(const float* __restrict__ Wtp, float* __restrict__ Bmat,
                               float* __restrict__ statsZero) {
    __shared__ float emb_s[27 * 4];
    __shared__ float y1_s[27 * 3];
    __shared__ float y2_s[27 * 5];
    __shared__ float w_s[27 * 160];
    int tid = threadIdx.x;

    if (tid < 27) {
        int i = tid / 9, j = (tid % 9) / 3, k = tid % 3;
        float cx = (float)i - 1.0f, cy = (float)j - 1.0f, cz = (float)k - 1.0f;
        float nrm = sqrtf(cx*cx + cy*cy + cz*cz);
        float inv = 1.0f / fmaxf(nrm, 1e-12f);
        float ux = cx * inv, uy = cy * inv, uz = cz * inv;
        const float pref = 1.14136f * 7.38905609893065f; // 1.14136 * e^2
        #pragma unroll
        for (int r = 0; r < 4; ++r) {
            float v = (float)(r + 1) * 0.2f;
            float d = (nrm - v) * 5.0f;          // step = 0.2
            float t1 = d + 1.0f, t2 = 1.0f - d;
            float f1 = (t1 > 0.0f) ? expf(-1.0f / t1) : 0.0f;
            float f2 = (t2 > 0.0f) ? expf(-1.0f / t2) : 0.0f;
            emb_s[tid * 4 + r] = pref * f1 * f2;
        }
        const float s3 = 1.7320508075688772f, s15 = 3.872983346207417f, s5 = 2.23606797749979f;
        y1_s[tid*3+0] = s3*ux; y1_s[tid*3+1] = s3*uy; y1_s[tid*3+2] = s3*uz;
        y2_s[tid*5+0] = s15*ux*uy;
        y2_s[tid*5+1] = s15*uy*uz;
        y2_s[tid*5+2] = 0.5f*s5*(2.0f*uz*uz - ux*ux - uy*uy);
        y2_s[tid*5+3] = s15*ux*uz;
        y2_s[tid*5+4] = 0.5f*s15*(ux*ux - uy*uy);
    }
    __syncthreads();

    const float invK15 = 1.0f / 5.196152422706632f; // K^-1.5, K=3
    for (int idx = tid; idx < 27 * 160; idx += blockDim.x) {
        int p = idx / 160, q = idx % 160;
        float acc = 0.0f;
        #pragma unroll
        for (int r = 0; r < 4; ++r) acc += emb_s[p*4+r] * Wtp[r*160+q];
        w_s[idx] = acc * invK15;
    }
    __syncthreads();

    for (int idx = tid; idx < KTOT * CPAD; idx += blockDim.x) {
        int kk = idx / CPAD, c = idx % CPAD;
        int ci = kk / 27, p = kk % 27;
        float val = 0.0f;
        if (c < 28) {                       // scalar: w0[p][ci][c]
            val = w_s[p*160 + ci*28 + c];
        } else if (c < 52) {                // l=1: w1[p][ci][m1]*y1[comp]
            int j = c - 28, m1 = j / 3, cp = j % 3;
            val = w_s[p*160 + 112 + ci*8 + m1] * y1_s[p*3+cp];
        } else if (c < CMID) {              // l=2: w2[p][ci][m2]*y2[comp]
            int j = c - 52, m2 = j / 5, cp = j % 5;
            val = w_s[p*160 + 144 + ci*4 + m2] * y2_s[p*5+cp];
        }
        Bmat[idx] = 0.5f * val;             // 1/sqrt(CIN)
    }
    for (int idx = tid; idx < 2 * BATCH * CMID; idx += blockDim.x) statsZero[idx] = 0.0f;
}

// ---- shared conv micro-kernel: stage slab + kernel, run 27x5 f32 WMMAs --------
__device__ __forceinline__ void conv_row_wmma(
    const float* __restrict__ x, const float* __restrict__ Bmat,
    float* xs, float* Bs, int b, int z, int y,
    int tid, int wave, int nlo, int hi, v8f acc[5]) {
    // stage input slab with zero halo
    for (int idx = tid; idx < CIN * 738; idx += 160) {
        int ci = idx / 738, r1 = idx % 738;
        int zz = r1 / 246, r2 = r1 % 246;
        int yy = r2 / 82, xx = r2 % 82;
        int gz = z + zz - 1, gy = y + yy - 1, gx = xx - 1;
        float v = 0.0f;
        if (gz >= 0 && gz < S && gy >= 0 && gy < S && gx >= 0 && gx < S)
            v = x[((((size_t)b * CIN + ci) * S + gz) * S + gy) * S + gx];
        xs[idx] = v;
    }
    for (int idx = tid; idx < KTOT * CPAD; idx += 160) Bs[idx] = Bmat[idx];
    __syncthreads();

    const int sxm = wave * 16 + nlo;         // A row (voxel) base into slab
    #pragma unroll
    for (int nt = 0; nt < 5; ++nt) acc[nt] = (v8f){0,0,0,0,0,0,0,0};

    #pragma unroll
    for (int step = 0; step < 27; ++step) {
        const int k0 = step * 4;
        // A 16x4 f32 layout: low lane-half holds K={0,1}, high half K={2,3}
        v2f a;
        a.x = hi ? xs[aoffs(k0 + 2) + sxm] : xs[aoffs(k0 + 0) + sxm];
        a.y = hi ? xs[aoffs(k0 + 3) + sxm] : xs[aoffs(k0 + 1) + sxm];
        const int rb = (k0 + 2 * hi) * CPAD; // B rows {0,2}/{1,3} across lane halves
        #pragma unroll
        for (int nt = 0; nt < 5; ++nt) {
            v2f bv;
            bv.x = Bs[rb + nt * 16 + nlo];
            bv.y = Bs[rb + CPAD + nt * 16 + nlo];
            acc[nt] = __builtin_amdgcn_wmma_f32_16x16x4_f32(
                false, a, false, bv, (short)0, acc[nt], false, false);
        }
    }
}

// ---------------- kernel 1: conv pass #1 — per-(b,c) stats only ----------------
__global__ __launch_bounds__(160)
void conv_stats_k(const float* __restrict__ x, const float* __restrict__ Bmat,
                  float* __restrict__ gSum, float* __restrict__ gSS) {
    __shared__ float xs[CIN * 3 * 3 * 82];   // 2952
    __shared__ float Bs[KTOT * CPAD];        // 8640
    __shared__ float sdS[CMID];
    __shared__ float sdQ[CMID];

    const int tid = threadIdx.x;
    const int row = blockIdx.x;
    const int b = row / (S * S);
    const int rem = row % (S * S);
    const int z = rem / S, y = rem % S;
    const int wave = tid >> 5, lane = tid & 31, nlo = lane & 15, hi = lane >> 4;

    if (tid < CMID) { sdS[tid] = 0.0f; sdQ[tid] = 0.0f; }

    v8f acc[5];
    conv_row_wmma(x, Bmat, xs, Bs, b, z, y, tid, wave, nlo, hi, acc);

    #pragma unroll
    for (int nt = 0; nt < 5; ++nt) {
        const int c = nt * 16 + nlo;
        if (c < CMID) {
            float s = 0.0f, q = 0.0f;
            #pragma unroll
            for (int r = 0; r < 8; ++r) { float v = acc[nt][r]; s += v; q += v * v; }
            atomicAdd(&sdS[c], s);
            atomicAdd(&sdQ[c], q);
        }
    }
    __syncthreads();
    if (tid < CMID) {
        atomicAdd(&gSum[b * CMID + tid], sdS[tid]);
        atomicAdd(&gSS [b * CMID + tid], sdQ[tid]);
    }
}

// ---------------- kernel 2: finalize per-(b,c) affine scale/shift --------------
__global__ void finalize_k(const float* __restrict__ gSum, const float* __restrict__ gSS,
                           const float* __restrict__ bw, const float* __restrict__ bb,
                           float* __restrict__ scA, float* __restrict__ shA) {
    int t = threadIdx.x;
    if (t >= BATCH * CMID) return;
    int b = t / CMID, c = t % CMID;
    const float invV = 1.0f / (float)VOX;
    float sc, sh;
    if (c < 28) {
        float mean = gSum[t] * invV;
        float var  = gSS[t] * invV - mean * mean;
        float inv  = 1.0f / sqrtf(var + EPSV);
        sc = inv * bw[c];
        sh = bb[c] - mean * sc;
    } else if (c < 52) {
        int m1 = (c - 28) / 3, base = b * CMID + 28 + m1 * 3;
        float n1 = (gSS[base] + gSS[base+1] + gSS[base+2]) * (invV / 3.0f);
        sc = (1.0f / sqrtf(n1 + EPSV)) * bw[28 + m1];
        sh = 0.0f;
    } else {
        int m2 = (c - 52) / 5, base = b * CMID + 52 + m2 * 5;
        float n2 = (gSS[base] + gSS[base+1] + gSS[base+2] + gSS[base+3] + gSS[base+4]) * (invV / 5.0f);
        sc = (1.0f / sqrtf(n2 + EPSV)) * bw[36 + m2];
        sh = 0.0f;
    }
    scA[t] = sc; shA[t] = sh;
}

// ------- kernel 3: conv pass #2 (recompute) + norm + gate -> NCDHW out ---------
__global__ __launch_bounds__(160)
void conv_apply_k(const float* __restrict__ x, const float* __restrict__ Bmat,
                  const float* __restrict__ scA, const float* __restrict__ shA,
                  float* __restrict__ out) {
    __shared__ float xs[CIN * 3 * 3 * 82];   // 2952
    __shared__ float Bs[KTOT * CPAD];        // 8640; reused as h' tile (80 x 81)

    const int tid = threadIdx.x;
    const int row = blockIdx.x;
    const int b = row / (S * S);
    const int rem = row % (S * S);
    const int z = rem / S, y = rem % S;
    const int wave = tid >> 5, lane = tid & 31, nlo = lane & 15, hi = lane >> 4;

    v8f acc[5];
    conv_row_wmma(x, Bmat, xs, Bs, b, z, y, tid, wave, nlo, hi, acc);

    // per-lane norm coefficients for owned channels
    float scv[5], shv[5];
    #pragma unroll
    for (int nt = 0; nt < 5; ++nt) {
        const int c = nt * 16 + nlo;
        scv[nt] = (c < CMID) ? scA[b * CMID + c] : 0.0f;
        shv[nt] = (c < CMID) ? shA[b * CMID + c] : 0.0f;
    }

    __syncthreads();                         // done reading Bs -> reuse as tile
    float* hs = Bs;                          // h' tile: [80 voxels][stride 81]
    #pragma unroll
    for (int nt = 0; nt < 5; ++nt) {
        const int c = nt * 16 + nlo;
        if (c < CMID) {
            #pragma unroll
            for (int r = 0; r < 8; ++r) {
                const int xv = wave * 16 + r + 8 * hi;   // D row M = r + 8*hi
                hs[xv * 81 + c] = acc[nt][r] * scv[nt] + shv[nt];
            }
        }
    }
    __syncthreads();

    // gate + write NCDHW, coalesced 80-wide rows per output channel
    const size_t vrow = (((size_t)b * S + z) * S + y) * S;
    for (int idx = tid; idx < COUT * S; idx += 160) {
        const int co = idx / S, xv = idx % S;
        const float* hp = &hs[xv * 81];
        float o;
        if (co < NS_) {
            o = RELU_C * fmaxf(hp[co], 0.0f);
        } else if (co < 16 + 3 * M1_) {
            const int j = co - 16, m1 = j / 3;
            const float gv = SIG_C / (1.0f + expf(-hp[NS_ + m1]));
            o = hp[28 + j] * gv;
        } else {
            const int j = co - 40, m2 = j / 5;
            const float gv = SIG_C / (1.0f + expf(-hp[NS_ + M1_ + m2]));
            o = hp[52 + j] * gv;
        }
        out[((size_t)b * COUT + co) * VOX + vrow + xv] = o;
    }
}

extern "C" void kernel_launch(void* const* d_in, const int* in_sizes, int n_in,
                              void* d_out, int out_size, void* d_ws, size_t ws_size,
                              hipStream_t stream) {
    const float* x   = (const float*)d_in[0];
    const float* Wtp = (const float*)d_in[1];
    const float* bw  = (const float*)d_in[2];
    const float* bb  = (const float*)d_in[3];
    float* ws = (float*)d_ws;
    float* Bmat = ws + OFF_BMAT;
    float* gSum = ws + OFF_SUM;
    float* gSS  = ws + OFF_SS;
    float* scA  = ws + OFF_SC;
    float* shA  = ws + OFF_SH;

    build_kernel_k<<<1, 192, 0, stream>>>(Wtp, Bmat, gSum);
    conv_stats_k<<<BATCH * S * S, 160, 0, stream>>>(x, Bmat, gSum, gSS);
    finalize_k<<<1, BATCH * CMID, 0, stream>>>(gSum, gSS, bw, bb, scA, shA);
    conv_apply_k<<<BATCH * S * S, 160, 0, stream>>>(x, Bmat, scA, shA, (float*)d_out);
}